// NENN_27238682591290
// MI455X (gfx1250) — compile-verified
//
#include <hip/hip_runtime.h>
#include <math.h>

// ---------------------------------------------------------------------------
// NENN forward for MI455X (gfx1250, wave32).
//  * Ef: gather edge_features[src,dst] (4096 rows) then 32x32 project —
//    avoids the reference's full 1M-row GEMM (saves 134MB reads + 2.1 GFLOP).
//  * Three masked-softmax @ V products use V_WMMA_F32_16X16X4_F32 with a
//    two-pass *branchless* streaming softmax (scores are rank-1, recomputed).
//    Finite sentinel (-1e30) instead of -inf keeps the whole update chain in
//    v_max/v_exp/v_fma/v_cndmask — no EXEC divergence between WMMAs.
//  * Output cost is dominated by zero-filling the 403MB `res` tensor ->
//    float4 zero kernel + ordered sparse scatter ([src,dst] then [dst,src],
//    matching the reference's overwrite semantics).
// ---------------------------------------------------------------------------

typedef __attribute__((ext_vector_type(2))) float v2f;
typedef __attribute__((ext_vector_type(8))) float v8f;

#define NN 1024
#define EE 4096
#define DN 64
#define DE 32
#define SENT (-1.0e30f)

__device__ __forceinline__ float lrelu(float x) { return x >= 0.f ? x : 0.01f * x; }

// ---------------- Wn = node_features @ wnW.T + wnb  (1024x128 @ 128x64) ----
__global__ void k_wn(const float* __restrict__ nf, const float* __restrict__ W,
                     const float* __restrict__ b, float* __restrict__ Wn) {
  int i = blockIdx.x, j = threadIdx.x;      // 1024 blocks x 64 threads
  const float* row = nf + (size_t)i * 128;
  const float* w = W + (size_t)j * 128;
  float acc = b[j];
  for (int c = 0; c < 128; ++c) acc += row[c] * w[c];
  Wn[i * DN + j] = acc;
}

// ---------------- Ef = edge_features[src,dst] @ weW.T + web ----------------
__global__ void k_ef(const float* __restrict__ ef, const float* __restrict__ W,
                     const float* __restrict__ b, const int* __restrict__ srcI,
                     const int* __restrict__ dstI, float* __restrict__ Ef) {
  int e = blockIdx.x, j = threadIdx.x;      // 4096 blocks x 32 threads
  int s = srcI[e], d = dstI[e];
  const float* row = ef + ((size_t)s * NN + d) * DE;
  const float* w = W + (size_t)j * DE;
  float acc = b[j];
  for (int c = 0; c < DE; ++c) acc += row[c] * w[c];
  Ef[e * DE + j] = acc;
}

// ---------------- per-node scalar terms + degree ---------------------------
__global__ void k_node_terms(const float* __restrict__ Wn, const float* __restrict__ atnW,
                             const float* __restrict__ ateW, const float* __restrict__ qtnW,
                             const int* __restrict__ adj,
                             float* __restrict__ nA1, float* __restrict__ nA2,
                             float* __restrict__ nB2, float* __restrict__ qn,
                             int* __restrict__ deg) {
  int i = blockIdx.x * blockDim.x + threadIdx.x;
  if (i >= NN) return;
  const float* w = Wn + (size_t)i * DN;
  float a1 = 0.f, a2 = 0.f, b2 = 0.f, q = 0.f;
  for (int c = 0; c < DN; ++c) {
    float v = w[c];
    a1 += v * atnW[c];        // A1 = atnW[0:64]
    a2 += v * atnW[DN + c];   // A2 = atnW[64:128]
    b2 += v * ateW[DE + c];   // B2 = ateW[32:96]
    q  += v * qtnW[c];        // Q1 = qtnW[0:64]
  }
  nA1[i] = a1; nA2[i] = a2; nB2[i] = b2; qn[i] = q;
  int d = 0;
  const int* ar = adj + (size_t)i * NN;
  for (int j = 0; j < NN; ++j) d += (ar[j] > 0) ? 1 : 0;
  deg[i] = d;
}

// ---------------- per-edge scalar terms ------------------------------------
__global__ void k_edge_terms(const float* __restrict__ Ef, const float* __restrict__ ateW,
                             const float* __restrict__ qtnW, const float* __restrict__ qteW,
                             float* __restrict__ eB1, float* __restrict__ eQ2,
                             float* __restrict__ eC1, float* __restrict__ eC2) {
  int e = blockIdx.x * blockDim.x + threadIdx.x;
  if (e >= EE) return;
  const float* f = Ef + (size_t)e * DE;
  float b1 = 0.f, q2 = 0.f, c1 = 0.f, c2 = 0.f;
  for (int c = 0; c < DE; ++c) {
    float v = f[c];
    b1 += v * ateW[c];        // B1 = ateW[0:32]
    q2 += v * qtnW[DN + c];   // Q2 = qtnW[64:96]
    c1 += v * qteW[c];        // C1 = qteW[0:32]
    c2 += v * qteW[DE + c];   // C2 = qteW[32:64]
  }
  eB1[e] = b1; eQ2[e] = q2; eC1[e] = c1; eC2[e] = c2;
}

// ---------------------------------------------------------------------------
// Fused masked-softmax attention:  out = relu(softmax(lrelu(col[k]+row[i]+b),
// mask) @ V), one wave (32 lanes) per 16-row tile, D = NT*16 columns.
// Pass 1: branchless streaming masked max/sum (lane L and L+16 split K into
//         even/odd k, then combine the halves by a shuffle).
// Pass 2: build 16x4 fp32 A-tiles of probabilities in registers and issue
//         V_WMMA_F32_16X16X4_F32 against B = V[kk..kk+3, t*16..t*16+15].
// MODE 0: mask = (k < deg[i]) ; MODE 1: mask = (src[k]==i || dst[k]==i)
// MODE 2: mask = (maskmat[i*K+k] > 0)
// ---------------------------------------------------------------------------
template <int MODE, int NT>
__global__ __launch_bounds__(32)
void attn_softmax_wmma(const float* __restrict__ colterm, const float* __restrict__ rowterm,
                       const float* __restrict__ biasp, const float* __restrict__ V, int K,
                       const int* __restrict__ deg, const int* __restrict__ srcI,
                       const int* __restrict__ dstI, const int* __restrict__ maskmat,
                       float* __restrict__ out, int ldout) {
  const int lane  = threadIdx.x;   // 0..31
  const int rlane = lane & 15;     // row within tile / N column within tile
  const int khalf = lane >> 4;     // 0: K pair {0,1}; 1: K pair {2,3}
  const int grow  = blockIdx.x * 16 + rlane;
  const float bias = biasp[0];
  const float rt   = rowterm[grow];
  int degr = 0;
  if (MODE == 0) degr = deg[grow];
  const size_t mrowoff = (size_t)grow * (size_t)K;

  // ---- pass 1: branchless masked streaming max & sum over this half of K --
  float m = SENT, s = 0.f;
  for (int k = khalf; k < K; k += 2) {
    if (MODE == 2 && (((k - khalf) & 62) == 0))
      __builtin_prefetch(maskmat + mrowoff + k + 1024, 0, 1);
    bool ok;
    if (MODE == 0)      ok = (k < degr);
    else if (MODE == 1) ok = (srcI[k] == grow) || (dstI[k] == grow);
    else                ok = (maskmat[mrowoff + k] > 0);
    float sc  = lrelu(colterm[k] + rt + bias);
    float sce = ok ? sc : SENT;
    float mn  = fmaxf(m, sce);
    s = s * __expf(m - mn) + (ok ? __expf(sc - mn) : 0.f);
    m = mn;
  }
  // combine the two half-lanes sharing a row (finite sentinel: no NaN cases)
  float mo = __shfl_xor(m, 16, 32);
  float so = __shfl_xor(s, 16, 32);
  float M = fmaxf(m, mo);
  float S = s * __expf(m - M) + so * __expf(mo - M);
  const float inv = 1.f / fmaxf(S, 1e-30f);

  // ---- pass 2: P @ V via f32 WMMA ----
  v8f acc[NT];
  const v8f zero8 = {0.f, 0.f, 0.f, 0.f, 0.f, 0.f, 0.f, 0.f};
#pragma unroll
  for (int t = 0; t < NT; ++t) acc[t] = zero8;

  const int D = NT * 16;
  for (int k0 = 0; k0 < K; k0 += 4) {
    const int kk = k0 + 2 * khalf;
    if (MODE == 2 && ((k0 & 124) == 0))
      __builtin_prefetch(maskmat + mrowoff + kk + 1024, 0, 1);
    v2f a;
#pragma unroll
    for (int j = 0; j < 2; ++j) {
      const int k = kk + j;
      bool ok;
      if (MODE == 0)      ok = (k < degr);
      else if (MODE == 1) ok = (srcI[k] == grow) || (dstI[k] == grow);
      else                ok = (maskmat[mrowoff + k] > 0);
      float sc = lrelu(colterm[k] + rt + bias);
      float pe = __expf(sc - M) * inv;   // finite; all-masked rows: ok==false
      a[j] = ok ? pe : 0.f;              // cndmask select, no divergence
    }
#pragma unroll
    for (int t = 0; t < NT; ++t) {
      const float* vp = V + (size_t)kk * D + t * 16 + rlane;
      v2f b;
      b[0] = vp[0];
      b[1] = vp[D];
      acc[t] = __builtin_amdgcn_wmma_f32_16x16x4_f32(
          /*neg_a=*/false, a, /*neg_b=*/false, b,
          /*c_mod=*/(short)0, acc[t], /*reuse_a=*/false, /*reuse_b=*/false);
    }
  }

  // ---- store with relu; C/D layout: VGPR v -> M=v (lanes 0-15), M=v+8 ----
#pragma unroll
  for (int t = 0; t < NT; ++t) {
#pragma unroll
    for (int v = 0; v < 8; ++v) {
      const int mrow = (lane < 16) ? v : (v + 8);
      const int orow = blockIdx.x * 16 + mrow;
      out[(size_t)orow * ldout + t * 16 + rlane] = fmaxf(acc[t][v], 0.f);
    }
  }
}

// ---------------- e_ni: two-way softmax over {src,dst} ---------------------
__global__ void k_eni(const float* __restrict__ Wn, const float* __restrict__ qn,
                      const float* __restrict__ eQ2, const float* __restrict__ qbp,
                      const int* __restrict__ srcI, const int* __restrict__ dstI,
                      float* __restrict__ e_ni) {
  int e = blockIdx.x, j = threadIdx.x;      // 4096 blocks x 64 threads
  int s = srcI[e], d = dstI[e];
  float base = eQ2[e] + qbp[0];
  float s0 = lrelu(qn[s] + base);
  float s1 = lrelu(qn[d] + base);
  float m  = fmaxf(s0, s1);
  float a0 = __expf(s0 - m), a1 = __expf(s1 - m);
  float inv = 1.f / (a0 + a1);
  float v = (a0 * Wn[(size_t)s * DN + j] + a1 * Wn[(size_t)d * DN + j]) * inv;
  e_ni[(size_t)e * DN + j] = fmaxf(v, 0.f);
}

// ---------------- zero-fill (float4) ---------------------------------------
__global__ void k_zero4(float4* __restrict__ p, size_t n4) {
  size_t i = (size_t)blockIdx.x * blockDim.x + threadIdx.x;
  if (i < n4) p[i] = make_float4(0.f, 0.f, 0.f, 0.f);
}

// ---------------- sparse scatter of edge embeddings ------------------------
__global__ void k_scatter(const float* __restrict__ e_ni, const float* __restrict__ e_ei,
                          const int* __restrict__ srcI, const int* __restrict__ dstI,
                          float* __restrict__ res, int swap) {
  int e = blockIdx.x, c = threadIdx.x;      // 4096 blocks x 96 threads
  float v = (c < DN) ? e_ni[(size_t)e * DN + c] : e_ei[(size_t)e * DE + (c - DN)];
  int a = srcI[e], b = dstI[e];
  if (swap) { int t = a; a = b; b = t; }
  res[((size_t)a * NN + b) * 96 + c] = v;
}

extern "C" void kernel_launch(void* const* d_in, const int* in_sizes, int n_in,
                              void* d_out, int out_size, void* d_ws, size_t ws_size,
                              hipStream_t stream) {
  (void)in_sizes; (void)n_in; (void)out_size; (void)ws_size;
  const float* node_features = (const float*)d_in[0];
  const float* edge_features = (const float*)d_in[1];
  const float* wnW  = (const float*)d_in[2];
  const float* wnb  = (const float*)d_in[3];
  const float* weW  = (const float*)d_in[4];
  const float* web  = (const float*)d_in[5];
  const float* atnW = (const float*)d_in[6];
  const float* atnb = (const float*)d_in[7];
  const float* ateW = (const float*)d_in[8];
  const float* ateb = (const float*)d_in[9];
  const float* qtnW = (const float*)d_in[10];
  const float* qtnb = (const float*)d_in[11];
  const float* qteW = (const float*)d_in[12];
  const float* qteb = (const float*)d_in[13];
  const int* edge_index = (const int*)d_in[14];
  const int* adj        = (const int*)d_in[15];
  const int* line_adj   = (const int*)d_in[16];
  const int* srcI = edge_index;
  const int* dstI = edge_index + EE;

  // workspace layout (~2.4 MB of fp32)
  float* ws   = (float*)d_ws;
  float* Wn   = ws;                 // 1024*64
  float* Ef   = Wn + NN * DN;       // 4096*32
  float* nA1  = Ef + EE * DE;
  float* nA2  = nA1 + NN;
  float* nB2  = nA2 + NN;
  float* qn   = nB2 + NN;
  float* eB1  = qn + NN;
  float* eQ2  = eB1 + EE;
  float* eC1  = eQ2 + EE;
  float* eC2  = eC1 + EE;
  int*   deg  = (int*)(eC2 + EE);   // 1024 ints
  float* e_ni = (float*)(deg + NN); // 4096*64
  float* e_ei = e_ni + EE * DN;     // 4096*32

  float* nodeEmb = (float*)d_out;         // [1024, 96]
  float* res = nodeEmb + (size_t)NN * 96; // [1024, 1024, 96]

  k_wn<<<NN, DN, 0, stream>>>(node_features, wnW, wnb, Wn);
  k_ef<<<EE, DE, 0, stream>>>(edge_features, weW, web, srcI, dstI, Ef);
  k_node_terms<<<NN / 256, 256, 0, stream>>>(Wn, atnW, ateW, qtnW, adj, nA1, nA2, nB2, qn, deg);
  k_edge_terms<<<EE / 256, 256, 0, stream>>>(Ef, ateW, qtnW, qteW, eB1, eQ2, eC1, eC2);

  // x_ni -> nodeEmb[:, 0:64]   (mask: j < deg[i])
  attn_softmax_wmma<0, 4><<<NN / 16, 32, 0, stream>>>(
      nA1, nA2, atnb, Wn, NN, deg, nullptr, nullptr, nullptr, nodeEmb, 96);
  // x_ei -> nodeEmb[:, 64:96]  (mask: src[k]==i || dst[k]==i)
  attn_softmax_wmma<1, 2><<<NN / 16, 32, 0, stream>>>(
      eB1, nB2, ateb, Ef, EE, nullptr, srcI, dstI, nullptr, nodeEmb + DN, 96);
  // e_ei -> workspace          (mask: line_adj > 0)
  attn_softmax_wmma<2, 2><<<EE / 16, 32, 0, stream>>>(
      eC1, eC2, qteb, Ef, EE, nullptr, nullptr, nullptr, line_adj, e_ei, DE);

  k_eni<<<EE, DN, 0, stream>>>(Wn, qn, eQ2, qtnb, srcI, dstI, e_ni);

  // res: zero 1024*1024*96 floats, then scatter (ordered: [src,dst] first,
  // then [dst,src], matching the reference's overwrite semantics)
  const size_t n4 = (size_t)NN * NN * 96 / 4;
  k_zero4<<<(unsigned)((n4 + 255) / 256), 256, 0, stream>>>((float4*)res, n4);
  k_scatter<<<EE, 96, 0, stream>>>(e_ni, e_ei, srcI, dstI, res, 0);
  k_scatter<<<EE, 96, 0, stream>>>(e_ni, e_ei, srcI, dstI, res, 1);
}